// NodeNet_1675037245679
// MI455X (gfx1250) — compile-verified
//
#include <hip/hip_runtime.h>
#include <hip/hip_bf16.h>
#include <math.h>

#define N_NODES 50000
#define N_EDGES 800000
#define X_DIM   128
#define HID     128
#define OUT_CH  128

typedef __attribute__((ext_vector_type(2))) float v2f;
typedef __attribute__((ext_vector_type(8))) float v8f;

// D = A(16x4 f32) * B(4x16 f32) + C(16x16 f32), full-precision fp32 WMMA.
__device__ __forceinline__ v8f wmma4(v2f a, v2f b, v8f c) {
    // (neg_a, A, neg_b, B, c_mod, C, reuse_a, reuse_b)
    return __builtin_amdgcn_wmma_f32_16x16x4_f32(false, a, false, b,
                                                 (short)0, c, false, false);
}

__device__ __forceinline__ float elu(float v) {
    return v > 0.0f ? v : expm1f(v);
}

// ---------------------------------------------------------------------------
// Stage 1: edge_sum[n, c] += edge_attr[e, c] for src node n = edge_index[0,e]
// One wave per edge; each lane handles 4 channels (float4 load + 4 f32 atomics).
// ---------------------------------------------------------------------------
__global__ __launch_bounds__(256)
void scatter_edges(const float* __restrict__ edge_attr,
                   const int*   __restrict__ edge_src,
                   float*       __restrict__ esum)
{
    long long tid = (long long)blockIdx.x * 256 + threadIdx.x;
    int e  = (int)(tid >> 5);
    int c4 = ((int)tid & 31) * 4;
    if (e >= N_EDGES) return;
    int node = edge_src[e];
    const float4 v = *(const float4*)(edge_attr + (long long)e * HID + c4);
    float* dst = esum + (long long)node * HID + c4;
    atomicAdd(dst + 0, v.x);
    atomicAdd(dst + 1, v.y);
    atomicAdd(dst + 2, v.z);
    atomicAdd(dst + 3, v.w);
}

// ---------------------------------------------------------------------------
// Stage 2: fused 3-layer MLP with fp32 WMMA.
// One wave per block; each wave computes a 32-row x 128-col output stripe as
// two 16-row M-tiles sharing B fragments. Layer outputs are staged through LDS
// to convert the C/D layout back into A-fragment layout (in-order DS ops, one
// wave -> no barrier needed).
// ---------------------------------------------------------------------------
__global__ __launch_bounds__(32)
void mlp_fused(const float* __restrict__ x,
               const float* __restrict__ esum,
               const float* __restrict__ W0, const float* __restrict__ b0,
               const float* __restrict__ W2, const float* __restrict__ b2,
               const float* __restrict__ W3, const float* __restrict__ b3,
               float* __restrict__ out)
{
    __shared__ float hbuf[2][16 * 128];   // 16 KB: per-M-tile staging

    const int lane = threadIdx.x;
    const int half = lane >> 4;           // 0: K 0/1 (or M 0..7), 1: K 2/3 (M 8..15)
    const int l15  = lane & 15;
    const int kb   = half * 2;            // K sub-offset inside a fragment

    int mA = blockIdx.x * 32;
    int mB = mA + 16;
    if (mB + 16 > N_NODES) mB = mA;       // last block: duplicate tile (same data)

    const v8f vzero = {0.f, 0.f, 0.f, 0.f, 0.f, 0.f, 0.f, 0.f};
    v8f accA[8], accB[8];
    #pragma unroll
    for (int t = 0; t < 8; ++t) { accA[t] = vzero; accB[t] = vzero; }

    // ---- Layer 0: [x | edge_sum] (K=256) @ W0 -------------------------------
    const float* rowAx = x + (size_t)(mA + l15) * X_DIM;
    const float* rowBx = x + (size_t)(mB + l15) * X_DIM;
    for (int k0 = 0; k0 < X_DIM; k0 += 4) {
        v2f a1 = *(const v2f*)(rowAx + k0 + kb);
        v2f a2 = *(const v2f*)(rowBx + k0 + kb);
        const float* wk = W0 + (size_t)(k0 + kb) * HID + l15;
        #pragma unroll
        for (int t = 0; t < 8; ++t) {
            v2f b; b.x = wk[t * 16]; b.y = wk[HID + t * 16];
            accA[t] = wmma4(a1, b, accA[t]);
            accB[t] = wmma4(a2, b, accB[t]);
        }
    }
    const float* rowAe = esum + (size_t)(mA + l15) * HID;
    const float* rowBe = esum + (size_t)(mB + l15) * HID;
    for (int k0 = 0; k0 < HID; k0 += 4) {
        v2f a1 = *(const v2f*)(rowAe + k0 + kb);
        v2f a2 = *(const v2f*)(rowBe + k0 + kb);
        const float* wk = W0 + (size_t)(X_DIM + k0 + kb) * HID + l15;
        #pragma unroll
        for (int t = 0; t < 8; ++t) {
            v2f b; b.x = wk[t * 16]; b.y = wk[HID + t * 16];
            accA[t] = wmma4(a1, b, accA[t]);
            accB[t] = wmma4(a2, b, accB[t]);
        }
    }
    // bias + ELU, stage to LDS in A-loadable (row-major) form
    #pragma unroll
    for (int t = 0; t < 8; ++t) {
        float bb = b0[t * 16 + l15];
        #pragma unroll
        for (int r = 0; r < 8; ++r) {
            int mrow = r + 8 * half;      // D layout: lanes16-31 hold M=8..15
            hbuf[0][mrow * 128 + t * 16 + l15] = elu(accA[t][r] + bb);
            hbuf[1][mrow * 128 + t * 16 + l15] = elu(accB[t][r] + bb);
        }
    }

    // ---- Layer 1: h (K=128) @ W2 -------------------------------------------
    #pragma unroll
    for (int t = 0; t < 8; ++t) { accA[t] = vzero; accB[t] = vzero; }
    for (int k0 = 0; k0 < HID; k0 += 4) {
        v2f a1 = *(const v2f*)&hbuf[0][l15 * 128 + k0 + kb];
        v2f a2 = *(const v2f*)&hbuf[1][l15 * 128 + k0 + kb];
        const float* wk = W2 + (size_t)(k0 + kb) * HID + l15;
        #pragma unroll
        for (int t = 0; t < 8; ++t) {
            v2f b; b.x = wk[t * 16]; b.y = wk[HID + t * 16];
            accA[t] = wmma4(a1, b, accA[t]);
            accB[t] = wmma4(a2, b, accB[t]);
        }
    }
    #pragma unroll
    for (int t = 0; t < 8; ++t) {
        float bb = b2[t * 16 + l15];
        #pragma unroll
        for (int r = 0; r < 8; ++r) {
            int mrow = r + 8 * half;
            hbuf[0][mrow * 128 + t * 16 + l15] = elu(accA[t][r] + bb);
            hbuf[1][mrow * 128 + t * 16 + l15] = elu(accB[t][r] + bb);
        }
    }

    // ---- Layer 2: h (K=128) @ W3, no activation ----------------------------
    #pragma unroll
    for (int t = 0; t < 8; ++t) { accA[t] = vzero; accB[t] = vzero; }
    for (int k0 = 0; k0 < HID; k0 += 4) {
        v2f a1 = *(const v2f*)&hbuf[0][l15 * 128 + k0 + kb];
        v2f a2 = *(const v2f*)&hbuf[1][l15 * 128 + k0 + kb];
        const float* wk = W3 + (size_t)(k0 + kb) * OUT_CH + l15;
        #pragma unroll
        for (int t = 0; t < 8; ++t) {
            v2f b; b.x = wk[t * 16]; b.y = wk[OUT_CH + t * 16];
            accA[t] = wmma4(a1, b, accA[t]);
            accB[t] = wmma4(a2, b, accB[t]);
        }
    }
    #pragma unroll
    for (int t = 0; t < 8; ++t) {
        float bb = b3[t * 16 + l15];
        #pragma unroll
        for (int r = 0; r < 8; ++r) {
            int mrow = r + 8 * half;
            out[(size_t)(mA + mrow) * OUT_CH + t * 16 + l15] = accA[t][r] + bb;
            out[(size_t)(mB + mrow) * OUT_CH + t * 16 + l15] = accB[t][r] + bb;
        }
    }
}

extern "C" void kernel_launch(void* const* d_in, const int* in_sizes, int n_in,
                              void* d_out, int out_size, void* d_ws, size_t ws_size,
                              hipStream_t stream) {
    const float* x   = (const float*)d_in[0];
    const int*   ei  = (const int*)  d_in[1];   // row 0 = source nodes
    const float* ea  = (const float*)d_in[2];
    const float* W0  = (const float*)d_in[3];
    const float* b0  = (const float*)d_in[4];
    const float* W2  = (const float*)d_in[5];
    const float* b2  = (const float*)d_in[6];
    const float* W3  = (const float*)d_in[7];
    const float* b3  = (const float*)d_in[8];
    float* out  = (float*)d_out;
    float* esum = (float*)d_ws;                 // N_NODES * HID floats (25.6 MB)

    hipMemsetAsync(esum, 0, (size_t)N_NODES * HID * sizeof(float), stream);

    // 800000 edges * 32 lanes / 256 threads = 100000 blocks
    scatter_edges<<<(N_EDGES * 32) / 256, 256, 0, stream>>>(ea, ei, esum);

    // one wave per 32 rows; 50000/32 -> 1563 blocks (last tile clamped)
    mlp_fused<<<(N_NODES + 31) / 32, 32, 0, stream>>>(x, esum,
                                                      W0, b0, W2, b2, W3, b3,
                                                      out);
}